// LightGraphConv_43198781063349
// MI455X (gfx1250) — compile-verified
//
#include <hip/hip_runtime.h>

// LightGCN propagation: out = diag(ci) * scatter_add(dst, gather(src, X*diag(cj)))
// N=100000 nodes, E=1.6M edges, D_FEAT=32 (== wave32 lane count).

#define D_FEAT 32

typedef float v2f __attribute__((ext_vector_type(2)));
typedef float v8f __attribute__((ext_vector_type(8)));

// ---------------- 1) zero the accumulator (must run every call: harness
// poisons d_out once and graph-replays without re-poisoning) ----------------
__global__ void lgc_zero_kernel(float4* __restrict__ p, int n4) {
  for (int i = blockIdx.x * blockDim.x + threadIdx.x; i < n4;
       i += gridDim.x * blockDim.x) {
    p[i] = make_float4(0.f, 0.f, 0.f, 0.f);
  }
}

// ---------------- 2) edge scatter: acc[dst] += feats[src] * cj[src] --------
// wave-per-edge: lane = feature (D_FEAT == 32 == wave32). Indices for 32
// edges are loaded coalesced once per batch and broadcast with __shfl.
__global__ void lgc_edge_scatter(const float* __restrict__ feats,
                                 const float* __restrict__ cj,
                                 const int* __restrict__ src,
                                 const int* __restrict__ dst,
                                 float* __restrict__ acc,
                                 int nEdges) {
  const int lane   = threadIdx.x & 31;
  const int wave   = (blockIdx.x * blockDim.x + threadIdx.x) >> 5;
  const int nWaves = (gridDim.x * blockDim.x) >> 5;

  for (int base = wave * 32; base < nEdges; base += nWaves * 32) {
    const int n = (nEdges - base < 32) ? (nEdges - base) : 32;
    int   sMy = 0, dMy = 0;
    float cMy = 0.f;
    if (lane < n) {
      sMy = src[base + lane];
      dMy = dst[base + lane];
      cMy = cj[sMy];
    }
    for (int i = 0; i < n; ++i) {
      const int   s = __shfl(sMy, i, 32);
      const int   d = __shfl(dMy, i, 32);
      const float c = __shfl(cMy, i, 32);
      const float v = feats[s * D_FEAT + lane] * c;   // coalesced 128B/edge, L2-resident
      unsafeAtomicAdd(&acc[d * D_FEAT + lane], v);     // hw global_atomic_add_f32, no return
    }
  }
}

// ---------------- 3) epilogue: out = diag(ci) * agg, via WMMA ---------------
// One wave per 16x16 tile (16 nodes x 16 features; D_FEAT=32 -> 2 col halves).
// O = D16 x Agg16 split along K into four K=4 chunks, chained through the
// f32 accumulator of V_WMMA_F32_16X16X4_F32.
//
// ISA layouts (CDNA5 ISA):
//   A 16x4 f32 (2 VGPRs): lanes0-15 M=l; VGPR0 holds K=2*hi, VGPR1 K=2*hi+1
//   B 4x16      (2 VGPRs): lanes hold N=l; VGPR0 row k0=2*hi, VGPR1 row k0+1
//   C/D 16x16   (8 VGPRs): VGPR v: M = v + 8*hi, N = l
__global__ void lgc_scale_wmma(float* __restrict__ out,
                               const float* __restrict__ ci, int nTiles) {
  const int wave = (blockIdx.x * blockDim.x + threadIdx.x) >> 5;
  if (wave >= nTiles) return;                 // wave-uniform: EXEC stays all-ones
  const int lane = threadIdx.x & 31;
  const int hi   = lane >> 4;
  const int l    = lane & 15;
  const int nodeBase = (wave >> 1) * 16;
  const int colBase  = (wave & 1) * 16;

  const float cv = ci[nodeBase + l];          // one diag value per M-row

  v8f c = {};
#pragma unroll
  for (int j = 0; j < 4; ++j) {
    const int k0 = 2 * hi;                    // K index held by VGPR0 in this lane half
    v2f a, b;
    a.x = (l == 4 * j + k0)     ? cv : 0.f;   // A[m][k] = diag slice, columns 4j..4j+3
    a.y = (l == 4 * j + k0 + 1) ? cv : 0.f;
    const float* rp = out + (nodeBase + 4 * j + k0) * D_FEAT + colBase + l;
    b.x = rp[0];                              // Agg row 4j+k0
    b.y = rp[D_FEAT];                         // Agg row 4j+k0+1
    c = __builtin_amdgcn_wmma_f32_16x16x4_f32(
        /*neg_a=*/false, a, /*neg_b=*/false, b,
        /*c_mod=*/(short)0, c, /*reuse_a=*/false, /*reuse_b=*/false);
  }
#pragma unroll
  for (int v = 0; v < 8; ++v) {
    out[(nodeBase + v + 8 * hi) * D_FEAT + colBase + l] = c[v];
  }
}

// Scalar tail for N % 16 nodes (N=100000 is divisible by 16; kept for safety).
__global__ void lgc_scale_tail(float* __restrict__ out,
                               const float* __restrict__ ci,
                               int startNode, int nNodes) {
  const int i = blockIdx.x * blockDim.x + threadIdx.x;
  const int total = nNodes * D_FEAT;
  if (i < total) {
    const int node = startNode + i / D_FEAT;
    out[node * D_FEAT + (i % D_FEAT)] *= ci[node];
  }
}

extern "C" void kernel_launch(void* const* d_in, const int* in_sizes, int n_in,
                              void* d_out, int out_size, void* d_ws, size_t ws_size,
                              hipStream_t stream) {
  const float* feats = (const float*)d_in[0];  // [N, 32]
  const float* cj    = (const float*)d_in[1];  // [N, 1]
  const float* ci    = (const float*)d_in[2];  // [N, 1]
  const int*   src   = (const int*)d_in[3];    // [E]
  const int*   dst   = (const int*)d_in[4];    // [E]
  float* out = (float*)d_out;                  // [N, 32]

  const int N = in_sizes[1];                   // cj: one entry per node
  const int E = in_sizes[3];

  // 1) zero accumulator (d_out doubles as the accumulator)
  const int n4 = (N * D_FEAT) / 4;
  int zb = (n4 + 255) / 256;
  if (zb > 2048) zb = 2048;
  lgc_zero_kernel<<<zb, 256, 0, stream>>>((float4*)out, n4);

  // 2) gather + atomic scatter-add over edges
  lgc_edge_scatter<<<2048, 256, 0, stream>>>(feats, cj, src, dst, out, E);

  // 3) row-scale by ci via WMMA tiles
  const int nTiles = (N / 16) * (D_FEAT / 16);
  if (nTiles > 0) {
    const int wavesPerBlock = 256 / 32;
    const int blocks = (nTiles + wavesPerBlock - 1) / wavesPerBlock;
    lgc_scale_wmma<<<blocks, 256, 0, stream>>>(out, ci, nTiles);
  }
  const int rem = N % 16;
  if (rem > 0) {
    const int t = rem * D_FEAT;
    lgc_scale_tail<<<(t + 255) / 256, 256, 0, stream>>>(out, ci, N - rem, rem);
  }
}